// DayAdapter_71622874628159
// MI455X (gfx1250) — compile-verified
//
#include <hip/hip_runtime.h>

typedef __attribute__((ext_vector_type(2))) float v2f;
typedef __attribute__((ext_vector_type(4))) float v4f;
typedef __attribute__((ext_vector_type(8))) float v8f;

#define DIM 512
#define TOK 1024
#define BM 128           // block rows   (4 wave-rows x 32)
#define BN 128           // block cols   (2 wave-cols x 64)
#define BK 32            // K chunk
#define XS_STRIDE 36     // padded row stride (floats): 16B-aligned stores, conflict-free b64 reads
#define WT_STRIDE 36     // transposed W tile: Wt[col][k], b64-aligned K pairs

static __device__ __forceinline__ v8f wmma4(v2f a, v2f b, v8f c) {
    return __builtin_amdgcn_wmma_f32_16x16x4_f32(
        false, a, false, b, (short)0, c, false, false);
}

__global__ __launch_bounds__(256)
void day_adapter_kernel(const float* __restrict__ x,
                        const int* __restrict__ day_ids,
                        const float* __restrict__ W,
                        const float* __restrict__ bias,
                        float* __restrict__ out) {
    __shared__ float Xs[2][BM * XS_STRIDE];   // 2 x 18 KB
    __shared__ float Wt[2][BN * WT_STRIDE];   // 2 x 18 KB (transposed)

    const int n0    = blockIdx.x * BN;
    const int m0    = blockIdx.y * BM;
    const int batch = blockIdx.z;
    const int day   = day_ids[batch];

    const int t    = threadIdx.x;
    const int lane = t & 31;
    const int wave = t >> 5;     // 8 waves: 4 M-rows x 2 N-cols
    const int mw   = wave >> 1;  // 0..3 -> 32-row strip
    const int nw   = wave & 1;   // 0..1 -> 64-col strip
    const int lm   = lane & 15;
    const int lg   = lane >> 4;

    const float* xb  = x + ((size_t)batch * TOK + m0) * DIM;
    const float* Wdn = W + (size_t)day * DIM * DIM + n0;

    const int arow0  = (mw << 5) + lm;        // A strip row (first 16x16 tile)
    const int arow1  = arow0 + 16;            // second 16x16 tile
    const int nwbase = nw << 6;               // wave's column base within block

    v8f acc[2][4];
    #pragma unroll
    for (int m = 0; m < 2; ++m)
        #pragma unroll
        for (int n = 0; n < 4; ++n) acc[m][n] = v8f{};

    auto load_tile = [&](int buf, int k0) {
        // X[128 x 32] row-major, padded stride
        #pragma unroll
        for (int i = 0; i < 4; ++i) {
            int f   = t + (i << 8);          // 1024 float4s
            int row = f >> 3;                // 8 float4 per row
            int c4  = (f & 7) << 2;
            v4f v = *(const v4f*)(xb + (size_t)row * DIM + (k0 + c4));
            *(v4f*)&Xs[buf][row * XS_STRIDE + c4] = v;
        }
        // W[32 x 128] -> transposed LDS Wt[col][k] so B frags are one b64 load
        #pragma unroll
        for (int i = 0; i < 4; ++i) {
            int f  = t + (i << 8);           // 1024 float4s
            int kr = f >> 5;                 // 32 float4 per k-row
            int c4 = (f & 31) << 2;
            v4f v = *(const v4f*)(Wdn + (size_t)(k0 + kr) * DIM + c4);
            #pragma unroll
            for (int j = 0; j < 4; ++j)
                Wt[buf][(c4 + j) * WT_STRIDE + kr] = v[j];
        }
    };

    load_tile(0, 0);
    __syncthreads();

    int cur = 0;
    for (int k0 = 0; k0 < DIM; k0 += BK) {
        const int nxt = cur ^ 1;
        if (k0 + BK < DIM) load_tile(nxt, k0 + BK);       // overlap with compute
        if (k0 + 2 * BK < DIM) {                          // prefetch chunk after next
            __builtin_prefetch(xb + (size_t)(t & 127) * DIM + (k0 + 2 * BK), 0, 1);
            __builtin_prefetch(Wdn + (size_t)(k0 + 2 * BK + (t >> 3)) * DIM + ((t & 7) << 4), 0, 1);
        }

        const float* xsc = &Xs[cur][0];
        const float* wtc = &Wt[cur][0];
        #pragma unroll
        for (int kk = 0; kk < BK; kk += 4) {
            const int kr = kk + (lg << 1);
            // A frags (16x4): lane holds M=lm, K = kr + {0,1}
            v2f a0 = *(const v2f*)&xsc[arow0 * XS_STRIDE + kr];
            v2f a1 = *(const v2f*)&xsc[arow1 * XS_STRIDE + kr];
            #pragma unroll
            for (int n = 0; n < 4; ++n) {
                // B frag (4x16): lane holds N=lm, K = kr + {0,1} — contiguous b64
                v2f bb = *(const v2f*)&wtc[(nwbase + (n << 4) + lm) * WT_STRIDE + kr];
                acc[0][n] = wmma4(a0, bb, acc[0][n]);
                acc[1][n] = wmma4(a1, bb, acc[1][n]);
            }
        }
        __syncthreads();
        cur = nxt;
    }

    // ---- epilogue: gathered bias + softsign + non-temporal store ----
    #pragma unroll
    for (int n = 0; n < 4; ++n) {
        const int col = n0 + nwbase + (n << 4) + lm;
        const float bv = bias[day * DIM + col];
        #pragma unroll
        for (int m = 0; m < 2; ++m) {
            #pragma unroll
            for (int r = 0; r < 8; ++r) {
                const int row = m0 + (mw << 5) + (m << 4) + r + (lg << 3);
                float y = acc[m][n][r] + bv;
                y = y / (1.0f + fabsf(y));
                __builtin_nontemporal_store(y, &out[((size_t)batch * TOK + row) * DIM + col]);
            }
        }
    }
}

extern "C" void kernel_launch(void* const* d_in, const int* in_sizes, int n_in,
                              void* d_out, int out_size, void* d_ws, size_t ws_size,
                              hipStream_t stream) {
    const float* x      = (const float*)d_in[0];
    const int* day_ids  = (const int*)d_in[1];
    const float* W      = (const float*)d_in[2];
    const float* b      = (const float*)d_in[3];
    float* out          = (float*)d_out;

    const int n_batch = in_sizes[1];  // 64
    dim3 grid(DIM / BN, TOK / BM, n_batch);
    day_adapter_kernel<<<grid, 256, 0, stream>>>(x, day_ids, W, b, out);
}